// SpatialEmbLoss_3d_59725815218268
// MI455X (gfx1250) — compile-verified
//
#include <hip/hip_runtime.h>
#include <hip/hip_bf16.h>

typedef __attribute__((ext_vector_type(2))) float v2f;
typedef __attribute__((ext_vector_type(8))) float v8f;

namespace {
constexpr int kB = 2, kC = 7, kD = 32, kH = 256, kW = 256;
constexpr int kVox = kD * kH * kW;            // 2,097,152
constexpr int kInst = 8;
constexpr int kBins = 1024;                   // Lovasz counting-sort bins over err in [0,2]
constexpr float kFgW = 10.0f;

// workspace layout (units of 4 bytes)
constexpr int OFF_TILES  = 0;                          // B * 16*16 WMMA accum tiles
constexpr int OFF_SEEDBG = OFF_TILES + kB * 256;       // B
constexpr int OFF_SEEDFG = OFF_SEEDBG + kB;            // B
constexpr int OFF_VARL   = OFF_SEEDFG + kB;            // B
constexpr int OFF_INSTL  = OFF_VARL + kB;              // B
constexpr int OFF_PAR    = OFF_INSTL + kB;             // B*8*8 (cx,cy,cz,s0,s1,s2,cnt,pad)
constexpr int OFF_FGH    = OFF_PAR + kB * kInst * 8;   // B*8*kBins u32
constexpr int OFF_BGH    = OFF_FGH + kB * kInst * kBins;
constexpr int WS_TOTAL   = OFF_BGH + kB * kInst * kBins;
} // namespace

__global__ void k_zero(float* __restrict__ ws, int n) {
  for (int k = blockIdx.x * blockDim.x + threadIdx.x; k < n; k += gridDim.x * blockDim.x)
    ws[k] = 0.0f;
}

// ---------------------------------------------------------------------------
// Pass 1: per-instance masked statistics as a 16x16xK fp32 GEMM on the WMMA
// pipe.  A rows (M): x,y,z, s0,s1,s2, 1, s0^2,s1^2,s2^2, 0...   B cols (N):
// mask(inst 1..8), mask&center(inst 1..8).  K = voxels, accumulated K=4 at a
// time with V_WMMA_F32_16X16X4_F32, tile atomically merged at the end.
// ---------------------------------------------------------------------------
__global__ void __launch_bounds__(256)
k_pass1(const float* __restrict__ pred, const int* __restrict__ inst,
        const unsigned char* __restrict__ cent, float* __restrict__ ws) {
  __shared__ float Q[8][32][17];   // [wave][voxel][row], padded to dodge bank conflicts
  __shared__ int   Iid[8][32];
  __shared__ int   Cfl[8][32];

  const int wave = threadIdx.x >> 5;
  const int lane = threadIdx.x & 31;
  const int r    = lane & 15;      // A row / B col
  const int hi   = lane >> 4;      // lane half selects K pair {0,1} vs {2,3}
  const int b    = blockIdx.y;

  const float*         predb = pred + (size_t)b * kC * kVox;
  const int*           instb = inst + (size_t)b * kVox;
  const unsigned char* cenb  = cent + (size_t)b * kVox;

  v8f acc = {0.f, 0.f, 0.f, 0.f, 0.f, 0.f, 0.f, 0.f};

  for (int base = blockIdx.x * 256; base < kVox; base += gridDim.x * 256) {
    const int v = base + (int)threadIdx.x;   // kVox % 256 == 0, always in range
    const float s0 = predb[(size_t)3 * kVox + v];
    const float s1 = predb[(size_t)4 * kVox + v];
    const float s2 = predb[(size_t)5 * kVox + v];
    const int   id = instb[v];
    const int   cf = cenb[v];
    const int   w  = v % kW;
    const int   h  = (v / kW) % kH;
    const int   d  = v / (kW * kH);
    const float x = w * (1.0f / (kW - 1));
    const float y = h * (1.0f / (kH - 1));
    const float z = d * (1.0f / (kD - 1));

    const float q[16] = {x, y, z, s0, s1, s2, 1.0f,
                         s0 * s0, s1 * s1, s2 * s2, 0, 0, 0, 0, 0, 0};
#pragma unroll
    for (int rr = 0; rr < 16; ++rr) Q[wave][lane][rr] = q[rr];
    Iid[wave][lane] = id;
    Cfl[wave][lane] = cf;
    __syncthreads();

#pragma unroll
    for (int j = 0; j < 8; ++j) {            // 8 x (K=4) = 32 voxels per wave stage
      const int v0 = j * 4 + hi * 2;
      const int v1 = v0 + 1;
      v2f a;
      a.x = Q[wave][v0][r];
      a.y = Q[wave][v1][r];
      const int i0 = Iid[wave][v0], i1 = Iid[wave][v1];
      const int c0 = Cfl[wave][v0], c1 = Cfl[wave][v1];
      v2f bb;
      bb.x = (r < 8) ? ((i0 == r + 1) ? 1.f : 0.f)
                     : ((i0 == r - 7 && c0) ? 1.f : 0.f);
      bb.y = (r < 8) ? ((i1 == r + 1) ? 1.f : 0.f)
                     : ((i1 == r - 7 && c1) ? 1.f : 0.f);
      acc = __builtin_amdgcn_wmma_f32_16x16x4_f32(false, a, false, bb,
                                                  (short)0, acc, false, false);
    }
    __syncthreads();
  }

  // merge 16x16 tile: lane holds D[m + hi*8][r] in acc[m]
  float* tile = ws + OFF_TILES + b * 256;
#pragma unroll
  for (int m = 0; m < 8; ++m) atomicAdd(&tile[(m + hi * 8) * 16 + r], acc[m]);
}

// ---------------------------------------------------------------------------
// Finalize per-instance parameters: center (pick vs mean), s=exp(10*mean_sigma)
// (nan_to_num), variance loss from the E[x^2]-2mE[x]+m^2*n identity.
// ---------------------------------------------------------------------------
__global__ void k_finalize(float* __restrict__ ws) {
  const int t = threadIdx.x;
  if (t >= kB * kInst) return;
  const int b = t >> 3, i = t & 7;
  const float* T = ws + OFF_TILES + b * 256;

  const float cnt   = T[6 * 16 + i];
  const float inv   = cnt > 0.f ? 1.f / cnt : 0.f;
  const float cmsum = T[6 * 16 + 8 + i];
  const bool  pick  = (cmsum == 1.0f);
  const float cx = pick ? T[0 * 16 + 8 + i] : T[0 * 16 + i] * inv;
  const float cy = pick ? T[1 * 16 + 8 + i] : T[1 * 16 + i] * inv;
  const float cz = pick ? T[2 * 16 + 8 + i] : T[2 * 16 + i] * inv;
  const float m0 = T[3 * 16 + i] * inv;
  const float m1 = T[4 * 16 + i] * inv;
  const float m2 = T[5 * 16 + i] * inv;

  auto nn = [](float s) {   // jnp.nan_to_num semantics
    if (!(s == s)) return 0.0f;
    if (s > 3.402823466e38f) return 3.402823466e38f;
    return s;
  };
  const float s0 = nn(__expf(10.f * m0));
  const float s1 = nn(__expf(10.f * m1));
  const float s2 = nn(__expf(10.f * m2));

  float var = 0.f;
  if (cnt > 0.f) {
    var = ((T[7 * 16 + i] - 2.f * m0 * T[3 * 16 + i] + m0 * m0 * cnt) +
           (T[8 * 16 + i] - 2.f * m1 * T[4 * 16 + i] + m1 * m1 * cnt) +
           (T[9 * 16 + i] - 2.f * m2 * T[5 * 16 + i] + m2 * m2 * cnt)) /
          (3.f * cnt);
  }
  atomicAdd(ws + OFF_VARL + b, var * (1.f / kInst));

  float* p = ws + OFF_PAR + (size_t)(b * kInst + i) * 8;
  p[0] = cx; p[1] = cy; p[2] = cz;
  p[3] = s0; p[4] = s1; p[5] = s2;
  p[6] = cnt; p[7] = 0.f;
}

// ---------------------------------------------------------------------------
// Pass 2: per-voxel dist per instance, seed losses, Lovasz error histograms
// (LDS sub-histograms, fg packed in upper 16 bits / bg in lower 16 bits,
// merged with global atomics).
// ---------------------------------------------------------------------------
__global__ void __launch_bounds__(256)
k_pass2(const float* __restrict__ pred, const int* __restrict__ inst,
        const unsigned char* __restrict__ labels, float* __restrict__ ws) {
  __shared__ unsigned int hist[kInst * kBins];   // 32 KB
  __shared__ float par[kInst][8];

  const int b = blockIdx.y;
  const float* parg = ws + OFF_PAR + (size_t)b * kInst * 8;
  for (int k = threadIdx.x; k < kInst * kBins; k += blockDim.x) hist[k] = 0u;
  if (threadIdx.x < kInst * 8) ((float*)par)[threadIdx.x] = parg[threadIdx.x];
  __syncthreads();

  const float*         predb = pred + (size_t)b * kC * kVox;
  const int*           instb = inst + (size_t)b * kVox;
  const unsigned char* labb  = labels + (size_t)b * kVox;

  float sbg = 0.f, sfg = 0.f;
  for (int v = blockIdx.x * blockDim.x + threadIdx.x; v < kVox;
       v += gridDim.x * blockDim.x) {
    const int w = v % kW, h = (v / kW) % kH, d = v / (kW * kH);
    const float e0 = tanhf(predb[v]) + w * (1.0f / (kW - 1));
    const float e1 = tanhf(predb[(size_t)kVox + v]) + h * (1.0f / (kH - 1));
    const float e2 = tanhf(predb[(size_t)2 * kVox + v]) + d * (1.0f / (kD - 1));
    const float seed = 1.f / (1.f + __expf(-predb[(size_t)6 * kVox + v]));
    const int id = instb[v];

#pragma unroll
    for (int i = 0; i < kInst; ++i) {
      const float dx = e0 - par[i][0];
      const float dy = e1 - par[i][1];
      const float dz = e2 - par[i][2];
      const float dist =
          __expf(-(dx * dx * par[i][3] + dy * dy * par[i][4] + dz * dz * par[i][5]));
      const bool fg = (id == i + 1);
      // errors = 1 - (2*dist-1)*sign  -> fg: 2-2*dist, bg: 2*dist, in [0,2]
      float err = fg ? (2.f - 2.f * dist) : (2.f * dist);
      int bin = (int)(err * (kBins * 0.5f));
      bin = bin < 0 ? 0 : (bin > kBins - 1 ? kBins - 1 : bin);
      atomicAdd(&hist[i * kBins + bin], fg ? 65536u : 1u);
      if (fg) sfg += kFgW * (seed - dist) * (seed - dist);
    }
    if (!labb[v]) sbg += seed * seed;
  }

  // wave32 reduction + one atomic per wave
  for (int o = 16; o > 0; o >>= 1) {
    sbg += __shfl_down(sbg, o, 32);
    sfg += __shfl_down(sfg, o, 32);
  }
  if ((threadIdx.x & 31) == 0) {
    atomicAdd(ws + OFF_SEEDBG + b, sbg);
    atomicAdd(ws + OFF_SEEDFG + b, sfg);
  }
  __syncthreads();

  unsigned int* fgH = (unsigned int*)ws + OFF_FGH + (size_t)b * kInst * kBins;
  unsigned int* bgH = (unsigned int*)ws + OFF_BGH + (size_t)b * kInst * kBins;
  for (int k = threadIdx.x; k < kInst * kBins; k += blockDim.x) {
    const unsigned int p = hist[k];
    if (p) {
      atomicAdd(&fgH[k], p >> 16);
      atomicAdd(&bgH[k], p & 0xFFFFu);
    }
  }
}

// ---------------------------------------------------------------------------
// Lovasz hinge via counting sort: descending bin scan; contributions telescope
// per equal-error run:  sum_bins e_bin * (F(after) - F(before)),
// F(cfg,cbg) = 1 - (gts-cfg)/(gts+cbg),  F(0,0) = 0 = jac_{-1}.
// ---------------------------------------------------------------------------
__global__ void k_lovasz(float* __restrict__ ws) {
  const int t = blockIdx.x * blockDim.x + threadIdx.x;
  if (t >= kB * kInst) return;
  const int b = t >> 3, i = t & 7;
  const unsigned int* f = (const unsigned int*)ws + OFF_FGH + (size_t)(b * kInst + i) * kBins;
  const unsigned int* g = (const unsigned int*)ws + OFF_BGH + (size_t)(b * kInst + i) * kBins;
  const float gts = ws[OFF_PAR + (size_t)(b * kInst + i) * 8 + 6];

  float loss = 0.f;
  if (gts > 0.f) {
    float cfg = 0.f, cbg = 0.f, jprev = 0.f;
    for (int bin = kBins - 1; bin >= 0; --bin) {
      const unsigned int fc = f[bin], bc = g[bin];
      if (fc | bc) {
        cfg += (float)fc;
        cbg += (float)bc;
        const float jac = 1.f - (gts - cfg) / (gts + cbg);
        const float e = (bin + 0.5f) * (2.0f / kBins);
        loss += e * (jac - jprev);
        jprev = jac;
      }
    }
  }
  atomicAdd(ws + OFF_INSTL + b, loss * (1.f / kInst));
}

__global__ void k_final(const float* __restrict__ ws, float* __restrict__ out) {
  if (threadIdx.x == 0 && blockIdx.x == 0) {
    float tot = 0.f;
    for (int b = 0; b < kB; ++b) {
      const float seed_loss =
          (ws[OFF_SEEDBG + b] + ws[OFF_SEEDFG + b]) * (1.0f / (float)kVox);
      tot += 1.0f * ws[OFF_INSTL + b] + 10.0f * ws[OFF_VARL + b] + 1.0f * seed_loss;
    }
    out[0] = tot * (1.0f / kB);
  }
}

extern "C" void kernel_launch(void* const* d_in, const int* in_sizes, int n_in,
                              void* d_out, int out_size, void* d_ws, size_t ws_size,
                              hipStream_t stream) {
  const float*         pred   = (const float*)d_in[0];
  const int*           inst   = (const int*)d_in[1];
  const unsigned char* labels = (const unsigned char*)d_in[2];
  const unsigned char* cents  = (const unsigned char*)d_in[3];
  float* ws  = (float*)d_ws;
  float* out = (float*)d_out;

  k_zero<<<256, 256, 0, stream>>>(ws, WS_TOTAL);

  dim3 g1(512, kB);
  k_pass1<<<g1, 256, 0, stream>>>(pred, inst, cents, ws);

  k_finalize<<<1, 32, 0, stream>>>(ws);

  dim3 g2(256, kB);
  k_pass2<<<g2, 256, 0, stream>>>(pred, inst, labels, ws);

  k_lovasz<<<1, 32, 0, stream>>>(ws);
  k_final<<<1, 32, 0, stream>>>(ws, out);
}